// NHAttention_89283780149548
// MI455X (gfx1250) — compile-verified
//
#include <hip/hip_runtime.h>

// ---------------------------------------------------------------------------
// MI455X / gfx1250 kernel.
//
// Reference semantics: out[b, n*C + o] = sum_c w_value[o,c] * x[b,c,n]
// (the whole attention branch is computed-then-discarded in the reference, so
// it is dead code w.r.t. d_out and is eliminated here).
//
// B=2, C=256, N=H*W=4096.  Memory-bound (~17 MB @ 23.3 TB/s).  fp32 kept
// end-to-end via V_WMMA_F32_16X16X4_F32 (the only f32 WMMA on CDNA5).
// ---------------------------------------------------------------------------

typedef __attribute__((ext_vector_type(2))) float v2f;
typedef __attribute__((ext_vector_type(8))) float v8f;

#define C_DIM 256
#define N_DIM 4096
#define NBATCH 2

// One wave (32 lanes) computes a 16(o) x 64(n) strip of value = W * X and
// stores it transposed into Out[b, n, o].
//
// WMMA operand layout (wave32, 16x16x4 f32):
//   A (16x4, W tile):  lane l: row m = l&15, ks = kb + 2*(l>>4) + {0,1}  (v2f)
//   B (4x16, X tile):  lane l: col n = l&15, ks = kb + 2*(l>>4) + {0,1}  (v2f)
//   C/D (16x16 f32):   lane l: col n = l&15, VGPR v: row o = v + 8*(l>>4)
// => per lane the 8 D components are 8 CONSECUTIVE floats of the transposed
//    output row, giving clean b128 stores.
__global__ __launch_bounds__(256)
void value_gemm_wmma_f32(const float* __restrict__ X,   // [B, C, N]
                         const float* __restrict__ W,   // [C, C] row-major [o][c]
                         float* __restrict__ Out)       // [B, N, C]
{
    const int lane = threadIdx.x & 31;
    const int wave = threadIdx.x >> 5;
    const int gw   = blockIdx.x * 8 + wave;          // 0..2047

    const int b      = gw >> 10;                     // batch: 0..1
    const int rem    = gw & 1023;
    const int o_base = (rem >> 6) << 4;              // 16 * o_tile   (0..240)
    const int n_base = (rem & 63) << 6;              // 64 * n_chunk  (0..4032)

    const int l15 = lane & 15;
    const int lhi = lane >> 4;                       // 0 or 1

    const float* Xb = X   + (size_t)b * C_DIM * N_DIM;
    float*       Ob = Out + (size_t)b * N_DIM * C_DIM;

    v8f acc0 = {}, acc1 = {}, acc2 = {}, acc3 = {};

    // A operand base: W[o_base + l15][2*lhi + kb .. +1]  (8B-aligned pair)
    const float* wrow = W  + (size_t)(o_base + l15) * C_DIM + 2 * lhi;
    // B operand base: X[2*lhi + kb][n_base + l15] (+N_DIM for second k)
    const float* xcol = Xb + (size_t)(2 * lhi) * N_DIM + n_base + l15;

    for (int kb = 0; kb < C_DIM; kb += 4) {
        v2f a = *(const v2f*)(wrow + kb);

        const float* xp = xcol + (size_t)kb * N_DIM;
        // Prefetch the next K-slab of X (gfx1250 global_prefetch path).
        __builtin_prefetch(xp + 4 * N_DIM, 0, 1);

        v2f b0, b1, b2, b3;
        b0.x = xp[0];   b0.y = xp[N_DIM];
        b1.x = xp[16];  b1.y = xp[N_DIM + 16];
        b2.x = xp[32];  b2.y = xp[N_DIM + 32];
        b3.x = xp[48];  b3.y = xp[N_DIM + 48];

        // 8-arg pattern: (neg_a, A, neg_b, B, c_mod, C, reuse_a, reuse_b)
        acc0 = __builtin_amdgcn_wmma_f32_16x16x4_f32(false, a, false, b0,
                                                     (short)0, acc0, false, false);
        acc1 = __builtin_amdgcn_wmma_f32_16x16x4_f32(false, a, false, b1,
                                                     (short)0, acc1, false, false);
        acc2 = __builtin_amdgcn_wmma_f32_16x16x4_f32(false, a, false, b2,
                                                     (short)0, acc2, false, false);
        acc3 = __builtin_amdgcn_wmma_f32_16x16x4_f32(false, a, false, b3,
                                                     (short)0, acc3, false, false);
    }

    // Transposed store: lane l writes Out[(n_base + 16*t + l15)*C + o_base + 8*lhi .. +7]
    // 32-byte aligned, 32-byte contiguous per lane -> b128 stores.
    float* p = Ob + (size_t)(n_base + l15) * C_DIM + o_base + 8 * lhi;
    *(v8f*)(p)                 = acc0;
    *(v8f*)(p + 16 * C_DIM)    = acc1;
    *(v8f*)(p + 32 * C_DIM)    = acc2;
    *(v8f*)(p + 48 * C_DIM)    = acc3;
}

extern "C" void kernel_launch(void* const* d_in, const int* in_sizes, int n_in,
                              void* d_out, int out_size, void* d_ws, size_t ws_size,
                              hipStream_t stream) {
    const float* x       = (const float*)d_in[0];  // [2,256,64,64]
    const float* w_value = (const float*)d_in[1];  // [256,256]
    // d_in[2..4] (w_qkv, w_proj, b_proj) feed only dead code in the reference.
    float* out = (float*)d_out;                    // [2,256,64,64] flat = [B,N,C]

    (void)in_sizes; (void)n_in; (void)out_size; (void)d_ws; (void)ws_size;

    // 2048 waves: 2 batches x 16 o-tiles x 64 n-chunks; 8 waves/block.
    value_gemm_wmma_f32<<<dim3(256), dim3(256), 0, stream>>>(x, w_value, out);
}